// RARNN_11467562680554
// MI455X (gfx1250) — compile-verified
//
#include <hip/hip_runtime.h>

// ---------------------------------------------------------------------------
// RARNN seq2seq (GRU encoder-decoder + attention) for MI455X / gfx1250.
//
//   1. gather_enc:        X[512,1024]  = enc_emb[word_inputs]
//   2. build_dec_input:   Xd[128,2048] = [ctx | dec_emb[tok_t]]
//   3. WMMA GEMM:         Gi_f/Gi_b = X @ enc_wih.T + bih   (f32 16x16x4, LDS
//                         double-buffered, ASYNCcnt-pipelined global->LDS)
//   4. WMMA GEMM:         Gd = Xd @ dec_wih.T + dec_bih
//   5. enc_recurrent:     512 sequential GRU steps, fwd+bwd (persistent WG)
//   6. combine:           enc_out = outs_f + outs_b ; dec_h = h_f + h_b
//   7. decoder_recurrent: 128 steps, attention, records Z=[h2|ctx]
//   8. WMMA GEMM:         logits = Z @ out_w.T + out_b  (16.8 GFLOP; out_w
//                         262 MB streamed <=2x from HBM behind async copies)
//   9. logsoftmax -> d_out ; attn rows written directly by decoder.
// ---------------------------------------------------------------------------

typedef __attribute__((ext_vector_type(2))) float v2f;
typedef __attribute__((ext_vector_type(8))) float v8f;
typedef __attribute__((ext_vector_type(4))) int   v4i;

#define AS1 __attribute__((address_space(1)))
#define AS3 __attribute__((address_space(3)))

#define Hh   1024
#define SRCN 512
#define TGTN 128
#define VOUT 32000

// GEMM tiling
#define BM   64
#define BN   128
#define KC   32
#define LDP  36        // padded LDS row stride (words): 16B-aligned, bank-conflict-free

#if __has_builtin(__builtin_amdgcn_global_load_async_to_lds_b128)
#define HAS_ASYNC_LDS 1
#else
#define HAS_ASYNC_LDS 0
#endif

__device__ __forceinline__ void copy16_g2l(const float* __restrict__ g, float* l) {
#if HAS_ASYNC_LDS
    __builtin_amdgcn_global_load_async_to_lds_b128(
        (AS1 v4i*)g, (AS3 v4i*)l, 0, 0);
#else
    *(float4*)l = *(const float4*)g;     // synchronous fallback
#endif
}

template <int N>
__device__ __forceinline__ void wait_async_le() {
#if HAS_ASYNC_LDS
#if __has_builtin(__builtin_amdgcn_s_wait_asynccnt)
    __builtin_amdgcn_s_wait_asynccnt((unsigned short)N);
#else
    asm volatile("s_wait_asynccnt %0" :: "i"(N) : "memory");
#endif
#endif
}

__device__ __forceinline__ float sigmoidf_(float x) {
    return 1.0f / (1.0f + __expf(-x));
}

__device__ __forceinline__ float wave_sum(float p) {
#pragma unroll
    for (int off = 16; off > 0; off >>= 1) p += __shfl_xor(p, off, 32);
    return p;
}

__device__ __forceinline__ float wave_max(float p) {
#pragma unroll
    for (int off = 16; off > 0; off >>= 1) p = fmaxf(p, __shfl_xor(p, off, 32));
    return p;
}

// ---------------------------------------------------------------------------
// WMMA f32 GEMM:  C[M,N] = A[M,K] @ W[N,K]^T + bias[N]
// Block: 256 threads = 8 waves; macro-tile BM x BN; wave tile 16(M) x 64(N).
// K consumed in KC-chunks staged through double-buffered LDS filled by
// async global->LDS b128 copies (ASYNCcnt), overlapped with WMMA compute.
// Requires M%64==0, N%128==0, K%32==0 (all shapes here satisfy this).
// ---------------------------------------------------------------------------
__global__ __launch_bounds__(256) void gemm_bias_wmma(
    const float* __restrict__ A, const float* __restrict__ W,
    const float* __restrict__ bias, float* __restrict__ C,
    int M, int N, int K)
{
    __shared__ float Als[2][BM * LDP];
    __shared__ float Bls[2][BN * LDP];

    const int tid  = threadIdx.x;
    const int Mb   = M / BM;
    const int mblk = blockIdx.x % Mb;
    const int nblk = blockIdx.x / Mb;
    const int m0b  = mblk * BM;
    const int n0b  = nblk * BN;

    const int wv   = tid >> 5;
    const int lane = tid & 31;
    const int mw   = wv & 3;             // 0..3 -> M sub-tile
    const int nw   = wv >> 2;            // 0..1 -> N half
    const int half = lane >> 4;          // ISA: lanes 16-31 hold K+2,K+3
    const int l16  = lane & 15;

    const int nch = K / KC;

    // ---- async fill of one KC-chunk (6 b128 copies per thread) ----
    auto issue = [&](int buf, int kc) {
        const int kbase = kc * KC;
#pragma unroll
        for (int i = 0; i < 2; ++i) {                  // A: 64 rows x 8 float4
            const int idx  = tid + 256 * i;
            const int row  = idx >> 3;
            const int colb = (idx & 7) << 2;
            copy16_g2l(A + (size_t)(m0b + row) * K + kbase + colb,
                       &Als[buf][row * LDP + colb]);
        }
#pragma unroll
        for (int i = 0; i < 4; ++i) {                  // B: 128 rows x 8 float4
            const int idx  = tid + 256 * i;
            const int row  = idx >> 3;
            const int colb = (idx & 7) << 2;
            copy16_g2l(W + (size_t)(n0b + row) * K + kbase + colb,
                       &Bls[buf][row * LDP + colb]);
        }
    };

    v8f c0 = {}, c1 = {}, c2 = {}, c3 = {};

    issue(0, 0);
    for (int kc = 0; kc < nch; ++kc) {
        const int cur = kc & 1;
        if (kc + 1 < nch) {
            issue(cur ^ 1, kc + 1);
            wait_async_le<6>();          // only the 6 just-issued may remain
        } else {
            wait_async_le<0>();
        }
        __syncthreads();                 // chunk kc visible to all waves

        const float* Ab = &Als[cur][(mw * 16 + l16) * LDP + 2 * half];
        const float* Bb = &Bls[cur][(nw * 64 + l16) * LDP + 2 * half];
#pragma unroll
        for (int ks = 0; ks < KC / 4; ++ks) {
            v2f a  = *(const v2f*)(Ab + ks * 4);
            v2f b0 = *(const v2f*)(Bb + 0 * 16 * LDP + ks * 4);
            v2f b1 = *(const v2f*)(Bb + 1 * 16 * LDP + ks * 4);
            v2f b2 = *(const v2f*)(Bb + 2 * 16 * LDP + ks * 4);
            v2f b3 = *(const v2f*)(Bb + 3 * 16 * LDP + ks * 4);
            c0 = __builtin_amdgcn_wmma_f32_16x16x4_f32(false, a, false, b0, (short)0, c0, false, false);
            c1 = __builtin_amdgcn_wmma_f32_16x16x4_f32(false, a, false, b1, (short)0, c1, false, false);
            c2 = __builtin_amdgcn_wmma_f32_16x16x4_f32(false, a, false, b2, (short)0, c2, false, false);
            c3 = __builtin_amdgcn_wmma_f32_16x16x4_f32(false, a, false, b3, (short)0, c3, false, false);
        }
        __syncthreads();                 // done reading cur before refill
    }

    const int m0 = m0b + mw * 16;
    const int n0 = n0b + nw * 64;
    const float bb0 = bias[n0 +  0 + l16];
    const float bb1 = bias[n0 + 16 + l16];
    const float bb2 = bias[n0 + 32 + l16];
    const float bb3 = bias[n0 + 48 + l16];
#pragma unroll
    for (int v = 0; v < 8; ++v) {
        const size_t mrow = (size_t)(m0 + half * 8 + v) * N;  // C/D: M = v + 8*half
        C[mrow + n0 +  0 + l16] = c0[v] + bb0;
        C[mrow + n0 + 16 + l16] = c1[v] + bb1;
        C[mrow + n0 + 32 + l16] = c2[v] + bb2;
        C[mrow + n0 + 48 + l16] = c3[v] + bb3;
    }
}

// ---------------------------------------------------------------------------
// Embedding gathers
// ---------------------------------------------------------------------------
__global__ __launch_bounds__(256) void gather_enc(
    const int* __restrict__ word_inputs, const float* __restrict__ enc_emb,
    float* __restrict__ X)
{
    const int i = blockIdx.x * 256 + threadIdx.x;
    if (i >= SRCN * Hh) return;
    const int t = i >> 10, j = i & (Hh - 1);
    X[i] = enc_emb[(size_t)word_inputs[t] * Hh + j];
}

__global__ __launch_bounds__(256) void build_dec_input(
    const int* __restrict__ ctx_idx, const int* __restrict__ targets,
    const float* __restrict__ rarnn_emb, const float* __restrict__ dec_emb,
    float* __restrict__ Xd)
{
    const int i = blockIdx.x * 256 + threadIdx.x;
    if (i >= TGTN * 2 * Hh) return;
    const int t = i >> 11, j = i & (2 * Hh - 1);
    float v;
    if (j < Hh) {
        v = rarnn_emb[(size_t)ctx_idx[0] * Hh + j];
    } else {
        const int tok = (t == 0) ? 0 : targets[t - 1];
        v = dec_emb[(size_t)tok * Hh + (j - Hh)];
    }
    Xd[i] = v;
}

__global__ __launch_bounds__(256) void combine_add(
    const float* __restrict__ a, const float* __restrict__ b,
    float* __restrict__ o, int n)
{
    const int i = blockIdx.x * 256 + threadIdx.x;
    if (i < n) o[i] = a[i] + b[i];
}

// ---------------------------------------------------------------------------
// Encoder recurrence: one persistent 1024-thread workgroup per direction.
// Wave-cooperative coalesced GEMV (32 waves x 96 rows, __shfl_xor reduction),
// then per-thread GRU gate math. Whh (12.6 MB) stays L2-resident (192 MB L2).
// ---------------------------------------------------------------------------
__global__ __launch_bounds__(1024) void enc_recurrent(
    const float* __restrict__ Gi_f, const float* __restrict__ Gi_b,
    const float* __restrict__ Whh_f, const float* __restrict__ Whh_b,
    const float* __restrict__ bhh_f, const float* __restrict__ bhh_b,
    float* __restrict__ outs_f, float* __restrict__ outs_b,
    float* __restrict__ hf, float* __restrict__ hb)
{
    const int dir = blockIdx.x;                 // 0 = forward, 1 = backward
    const float* Gi  = dir ? Gi_b  : Gi_f;
    const float* Whh = dir ? Whh_b : Whh_f;
    const float* bhh = dir ? bhh_b : bhh_f;
    float* outs = dir ? outs_b : outs_f;
    float* hfin = dir ? hb : hf;

    __shared__ float h[Hh];
    __shared__ float gh[3 * Hh];
    const int tid = threadIdx.x;
    const int wave = tid >> 5, lane = tid & 31;

    h[tid] = 0.0f;
    __syncthreads();

    for (int s = 0; s < SRCN; ++s) {
        const int t = dir ? (SRCN - 1 - s) : s;
        for (int r = wave; r < 3 * Hh; r += 32) {
            const float* wrow = Whh + (size_t)r * Hh;
            float p = 0.0f;
            for (int k = lane; k < Hh; k += 32) p = fmaf(wrow[k], h[k], p);
            p = wave_sum(p);
            if (lane == 0) gh[r] = p + bhh[r];
        }
        __syncthreads();

        const float* gi = Gi + (size_t)t * (3 * Hh);
        const float r_ = sigmoidf_(gi[tid]          + gh[tid]);
        const float z_ = sigmoidf_(gi[tid + Hh]     + gh[tid + Hh]);
        const float n_ = tanhf   (gi[tid + 2 * Hh] + r_ * gh[tid + 2 * Hh]);
        const float hnew = (1.0f - z_) * n_ + z_ * h[tid];
        h[tid] = hnew;                               // only own slot touched here
        outs[(size_t)t * Hh + tid] = hnew;
        __syncthreads();
    }
    hfin[tid] = h[tid];
}

// ---------------------------------------------------------------------------
// Decoder recurrence + attention (single persistent workgroup, 128 steps).
// Records z_t = [h2 | context] into Z so the 32000-way projection runs as one
// batched WMMA GEMM afterwards. Writes attn rows straight into d_out.
// ---------------------------------------------------------------------------
__global__ __launch_bounds__(1024) void decoder_recurrent(
    const float* __restrict__ Gd, const float* __restrict__ Whh,
    const float* __restrict__ bhh, const float* __restrict__ enc_out,
    const float* __restrict__ dec_h0,
    float* __restrict__ Z, float* __restrict__ attn_out)
{
    __shared__ float h[Hh];
    __shared__ float gh[3 * Hh];
    __shared__ float attn[SRCN];
    const int tid = threadIdx.x;
    const int wave = tid >> 5, lane = tid & 31;

    h[tid] = dec_h0[tid];
    __syncthreads();

    for (int t = 0; t < TGTN; ++t) {
        // gh = dec_whh @ h + bhh
        for (int r = wave; r < 3 * Hh; r += 32) {
            const float* wrow = Whh + (size_t)r * Hh;
            float p = 0.0f;
            for (int k = lane; k < Hh; k += 32) p = fmaf(wrow[k], h[k], p);
            p = wave_sum(p);
            if (lane == 0) gh[r] = p + bhh[r];
        }
        __syncthreads();

        const float* gi = Gd + (size_t)t * (3 * Hh);
        const float r_ = sigmoidf_(gi[tid]          + gh[tid]);
        const float z_ = sigmoidf_(gi[tid + Hh]     + gh[tid + Hh]);
        const float n_ = tanhf   (gi[tid + 2 * Hh] + r_ * gh[tid + 2 * Hh]);
        const float h2 = (1.0f - z_) * n_ + z_ * h[tid];
        h[tid] = h2;
        __syncthreads();

        // attn[s] = sigmoid(enc_out[s] . h2)   (wave-cooperative dots)
        for (int s2 = wave; s2 < SRCN; s2 += 32) {
            const float* er = enc_out + (size_t)s2 * Hh;
            float p = 0.0f;
            for (int k = lane; k < Hh; k += 32) p = fmaf(er[k], h[k], p);
            p = wave_sum(p);
            if (lane == 0) attn[s2] = sigmoidf_(p);
        }
        __syncthreads();

        if (t < TGTN - 1 && tid < SRCN)
            attn_out[(size_t)t * SRCN + tid] = attn[tid];

        // context[j] = sum_s attn[s] * enc_out[s][j]   (coalesced across j)
        float c = 0.0f;
        for (int s2 = 0; s2 < SRCN; ++s2)
            c = fmaf(attn[s2], enc_out[(size_t)s2 * Hh + tid], c);

        Z[(size_t)t * (2 * Hh) + tid]      = h[tid];
        Z[(size_t)t * (2 * Hh) + Hh + tid] = c;
        __syncthreads();
    }
}

// ---------------------------------------------------------------------------
// Row-wise log_softmax over 32000 logits, one block per row.
// ---------------------------------------------------------------------------
__global__ __launch_bounds__(256) void logsoftmax_kernel(
    const float* __restrict__ logits, float* __restrict__ out, int ncols)
{
    const int row = blockIdx.x;
    const float* x = logits + (size_t)row * ncols;
    float* o = out + (size_t)row * ncols;
    __shared__ float sred[8];
    __shared__ float sbc;
    const int tid = threadIdx.x, lane = tid & 31, wave = tid >> 5;

    float m = -3.4e38f;
    for (int i = tid; i < ncols; i += 256) m = fmaxf(m, x[i]);
    m = wave_max(m);
    if (lane == 0) sred[wave] = m;
    __syncthreads();
    if (wave == 0) {
        float mm = (lane < 8) ? sred[lane] : -3.4e38f;
        mm = wave_max(mm);
        if (lane == 0) sbc = mm;
    }
    __syncthreads();
    const float M = sbc;

    float s = 0.0f;
    for (int i = tid; i < ncols; i += 256) s += __expf(x[i] - M);
    s = wave_sum(s);
    if (lane == 0) sred[wave] = s;
    __syncthreads();
    if (wave == 0) {
        float ss = (lane < 8) ? sred[lane] : 0.0f;
        ss = wave_sum(ss);
        if (lane == 0) sbc = ss;
    }
    __syncthreads();
    const float lse = M + logf(sbc);
    for (int i = tid; i < ncols; i += 256) o[i] = x[i] - lse;
}

// ---------------------------------------------------------------------------
static inline void launch_gemm(const float* A, const float* W, const float* bias,
                               float* C, int M, int N, int K, hipStream_t s)
{
    const int blocks = (M / BM) * (N / BN);
    gemm_bias_wmma<<<blocks, 256, 0, s>>>(A, W, bias, C, M, N, K);
}

extern "C" void kernel_launch(void* const* d_in, const int* in_sizes, int n_in,
                              void* d_out, int out_size, void* d_ws, size_t ws_size,
                              hipStream_t stream) {
    const int*   context_input = (const int*)  d_in[0];
    const int*   word_inputs   = (const int*)  d_in[1];
    const int*   word_targets  = (const int*)  d_in[2];
    const float* rarnn_emb     = (const float*)d_in[3];
    const float* enc_emb       = (const float*)d_in[4];
    const float* enc_wih_f     = (const float*)d_in[5];
    const float* enc_whh_f     = (const float*)d_in[6];
    const float* enc_bih_f     = (const float*)d_in[7];
    const float* enc_bhh_f     = (const float*)d_in[8];
    const float* enc_wih_b     = (const float*)d_in[9];
    const float* enc_whh_b     = (const float*)d_in[10];
    const float* enc_bih_b     = (const float*)d_in[11];
    const float* enc_bhh_b     = (const float*)d_in[12];
    const float* dec_emb       = (const float*)d_in[13];
    const float* dec_wih       = (const float*)d_in[14];
    const float* dec_whh       = (const float*)d_in[15];
    const float* dec_bih       = (const float*)d_in[16];
    const float* dec_bhh       = (const float*)d_in[17];
    const float* out_w         = (const float*)d_in[18];
    const float* out_b         = (const float*)d_in[19];

    // Workspace layout (floats). Total ~10.26M floats ~= 41 MB.
    float* ws = (float*)d_ws;
    size_t o = 0;
    float* X      = ws + o; o += (size_t)SRCN * Hh;          // 512x1024
    float* Gi_f   = ws + o; o += (size_t)SRCN * 3 * Hh;      // 512x3072
    float* Gi_b   = ws + o; o += (size_t)SRCN * 3 * Hh;
    float* outs_f = ws + o; o += (size_t)SRCN * Hh;
    float* outs_b = ws + o; o += (size_t)SRCN * Hh;
    float* enc_o  = ws + o; o += (size_t)SRCN * Hh;
    float* Xd     = ws + o; o += (size_t)TGTN * 2 * Hh;      // 128x2048
    float* Gd     = ws + o; o += (size_t)TGTN * 3 * Hh;      // 128x3072
    float* hf     = ws + o; o += Hh;
    float* hb     = ws + o; o += Hh;
    float* dec_h0 = ws + o; o += Hh;
    float* Z      = ws + o; o += (size_t)TGTN * 2 * Hh;      // 128x2048
    float* logits = ws + o; o += (size_t)TGTN * VOUT;        // 128x32000

    float* out_lsm  = (float*)d_out;                          // 128 x 32000
    float* out_attn = (float*)d_out + (size_t)TGTN * VOUT;    // 127 x 512

    // 1-2: gathers
    gather_enc<<<(SRCN * Hh + 255) / 256, 256, 0, stream>>>(word_inputs, enc_emb, X);
    build_dec_input<<<(TGTN * 2 * Hh + 255) / 256, 256, 0, stream>>>(
        context_input, word_targets, rarnn_emb, dec_emb, Xd);

    // 3-4: batched input-side GEMMs (WMMA f32)
    launch_gemm(X,  enc_wih_f, enc_bih_f, Gi_f, SRCN, 3 * Hh, Hh,     stream);
    launch_gemm(X,  enc_wih_b, enc_bih_b, Gi_b, SRCN, 3 * Hh, Hh,     stream);
    launch_gemm(Xd, dec_wih,   dec_bih,   Gd,   TGTN, 3 * Hh, 2 * Hh, stream);

    // 5: bidirectional encoder recurrence (2 persistent workgroups)
    enc_recurrent<<<2, 1024, 0, stream>>>(Gi_f, Gi_b, enc_whh_f, enc_whh_b,
                                          enc_bhh_f, enc_bhh_b,
                                          outs_f, outs_b, hf, hb);

    // 6: combine
    combine_add<<<(SRCN * Hh + 255) / 256, 256, 0, stream>>>(outs_f, outs_b, enc_o, SRCN * Hh);
    combine_add<<<(Hh + 255) / 256, 256, 0, stream>>>(hf, hb, dec_h0, Hh);

    // 7: decoder recurrence + attention
    decoder_recurrent<<<1, 1024, 0, stream>>>(Gd, dec_whh, dec_bhh, enc_o, dec_h0,
                                              Z, out_attn);

    // 8: output projection (dominant GEMM: 128x2048 @ 2048x32000, WMMA f32)
    launch_gemm(Z, out_w, out_b, logits, TGTN, VOUT, 2 * Hh, stream);

    // 9: log_softmax into d_out
    logsoftmax_kernel<<<TGTN, 256, 0, stream>>>(logits, out_lsm, VOUT);
}